// SSMRecurrentBlock_4389456577367
// MI455X (gfx1250) — compile-verified
//
#include <hip/hip_runtime.h>
#include <hip/hip_bf16.h>

#define D_MODEL 2048
#define D_INNER 4096
#define N_STATE 16
#define BATCH   128

typedef __attribute__((ext_vector_type(2))) float v2f;
typedef __attribute__((ext_vector_type(8))) float v8f;

// One 16x16 fp32 output tile accumulated over K with V_WMMA_F32_16X16X4_F32.
// ap/bp are per-lane base pointers already adjusted by +2*(lane>=16) so each
// K-step of 4 is a single contiguous float2 (global_load_b64) per operand.
__device__ __forceinline__ v8f gemm_tile_f32(const float* __restrict__ ap,
                                             const float* __restrict__ bp,
                                             int K, v8f acc) {
    for (int k = 0; k < K; k += 16) {
        // keep the weight stream ahead of consumption (global_prefetch_b8)
        __builtin_prefetch(bp + k + 1024, 0, 1);
        #pragma unroll
        for (int kk = 0; kk < 16; kk += 4) {
            v2f a = *(const v2f*)(ap + k + kk);
            v2f b = *(const v2f*)(bp + k + kk);
            acc = __builtin_amdgcn_wmma_f32_16x16x4_f32(
                false, a, false, b, (short)0, acc, false, false);
        }
    }
    return acc;
}

// Fused: x_proj, B/C projections, state update, readout accumulation.
// grid = D_INNER/16 = 256 blocks; block = 256 threads = 8 waves.
// Wave w handles rows [16w, 16w+16); block jt handles d_inner cols [16jt, 16jt+16).
__global__ __launch_bounds__(256) void ssm_fused_kernel(
    const float* __restrict__ x,      // [B, D_MODEL]
    const float* __restrict__ h,      // [B, N_STATE, D_INNER]
    const float* __restrict__ A,      // [N_STATE]
    const float* __restrict__ Dp,     // [D_INNER]
    const float* __restrict__ W_in,   // [D_INNER, D_MODEL]
    const float* __restrict__ b_in,   // [D_INNER]
    const float* __restrict__ W_B,    // [N_STATE*D_INNER, D_MODEL]
    const float* __restrict__ b_B,
    const float* __restrict__ W_C,
    const float* __restrict__ b_C,
    float* __restrict__ h_out,        // [B, N_STATE, D_INNER]
    float* __restrict__ y_ws)         // [B, D_INNER]
{
    const int lane = threadIdx.x & 31;
    const int wave = threadIdx.x >> 5;
    const int half = lane >> 4;       // 0: K/M rows 0..; 1: +2 K offset, rows +8
    const int ln16 = lane & 15;       // A: M index; B: N index
    const int j0   = blockIdx.x * 16; // d_inner column tile
    const int r0   = wave * 16;       // batch row tile
    const int j    = j0 + ln16;

    // per-lane operand base pointers
    const float* ap = x + (size_t)(r0 + ln16) * D_MODEL + 2 * half;

    // ---- x_proj tile: x @ W_in.T + b_in ----
    v8f xp = {};
    xp = gemm_tile_f32(ap, W_in + (size_t)j * D_MODEL + 2 * half, D_MODEL, xp);
    const float bin = b_in[j];
    #pragma unroll
    for (int v = 0; v < 8; ++v) xp[v] += bin;

    // readout accumulator starts at Dp * x_proj
    const float dpv = Dp[j];
    v8f y;
    #pragma unroll
    for (int v = 0; v < 8; ++v) y[v] = dpv * xp[v];

    // ---- per-state: Bm, Cm projections fused with state update ----
    for (int s = 0; s < N_STATE; ++s) {
        const float sigA = 1.0f / (1.0f + __expf(-A[s]));
        const size_t wrow = (size_t)(s * D_INNER + j);
        v8f accB = {};
        accB = gemm_tile_f32(ap, W_B + wrow * D_MODEL + 2 * half, D_MODEL, accB);
        v8f accC = {};
        accC = gemm_tile_f32(ap, W_C + wrow * D_MODEL + 2 * half, D_MODEL, accC);
        const float bB = b_B[wrow];
        const float bC = b_C[wrow];
        #pragma unroll
        for (int v = 0; v < 8; ++v) {
            const int row = r0 + v + 8 * half;                 // C/D tile row
            const size_t hidx = ((size_t)row * N_STATE + s) * D_INNER + j;
            const float hn = sigA * h[hidx] + (accB[v] + bB) * xp[v];
            h_out[hidx] = hn;
            y[v] += (accC[v] + bC) * hn;
        }
    }

    // stage y for the output GEMM
    #pragma unroll
    for (int v = 0; v < 8; ++v) {
        const int row = r0 + v + 8 * half;
        y_ws[(size_t)row * D_INNER + j] = y[v];
    }
}

// output[B, D_MODEL] = y[B, D_INNER] @ W_out.T + b_out
// grid = D_MODEL/16 = 128 blocks; 8 waves each own an M tile.
__global__ __launch_bounds__(256) void out_gemm_kernel(
    const float* __restrict__ y,      // [B, D_INNER]
    const float* __restrict__ W_out,  // [D_MODEL, D_INNER]
    const float* __restrict__ b_out,  // [D_MODEL]
    float* __restrict__ out)          // [B, D_MODEL]
{
    const int lane = threadIdx.x & 31;
    const int wave = threadIdx.x >> 5;
    const int half = lane >> 4;
    const int ln16 = lane & 15;
    const int j0   = blockIdx.x * 16;
    const int r0   = wave * 16;
    const int j    = j0 + ln16;

    const float* ap = y     + (size_t)(r0 + ln16) * D_INNER + 2 * half;
    const float* bp = W_out + (size_t)j * D_INNER + 2 * half;

    v8f acc = {};
    acc = gemm_tile_f32(ap, bp, D_INNER, acc);

    const float bo = b_out[j];
    #pragma unroll
    for (int v = 0; v < 8; ++v) {
        const int row = r0 + v + 8 * half;
        out[(size_t)row * D_MODEL + j] = acc[v] + bo;
    }
}

extern "C" void kernel_launch(void* const* d_in, const int* in_sizes, int n_in,
                              void* d_out, int out_size, void* d_ws, size_t ws_size,
                              hipStream_t stream) {
    (void)in_sizes; (void)n_in; (void)out_size; (void)ws_size;

    const float* x     = (const float*)d_in[0];
    const float* h     = (const float*)d_in[1];
    const float* A     = (const float*)d_in[2];
    const float* Dp    = (const float*)d_in[3];
    const float* W_in  = (const float*)d_in[4];
    const float* b_in  = (const float*)d_in[5];
    const float* W_B   = (const float*)d_in[6];
    const float* b_B   = (const float*)d_in[7];
    const float* W_C   = (const float*)d_in[8];
    const float* b_C   = (const float*)d_in[9];
    const float* W_out = (const float*)d_in[10];
    const float* b_out = (const float*)d_in[11];

    float* out   = (float*)d_out;                         // [128, 2048]
    float* h_out = out + (size_t)BATCH * D_MODEL;         // [128, 16, 4096]
    float* y_ws  = (float*)d_ws;                          // [128, 4096] = 2 MB

    ssm_fused_kernel<<<D_INNER / 16, 256, 0, stream>>>(
        x, h, A, Dp, W_in, b_in, W_B, b_B, W_C, b_C, h_out, y_ws);
    out_gemm_kernel<<<D_MODEL / 16, 256, 0, stream>>>(
        y_ws, W_out, b_out, out);
}